// ElementLevelScatterModuleDict_62148176773688
// MI455X (gfx1250) — compile-verified
//
#include <hip/hip_runtime.h>
#include <cstddef>

typedef float v2f __attribute__((ext_vector_type(2)));
typedef float v8f __attribute__((ext_vector_type(8)));

#define TM 64        // edges (M rows) per block
#define TPB 256      // 8 wave32 waves
#define LDA 132      // padded LDS row stride (floats): conflict-free ds_load_b64

// ---------------------------------------------------------------------------
// Zero d_out and the per-node count scratch (outputs are accumulated with
// atomics, and the harness poisons buffers with 0xAA).
// ---------------------------------------------------------------------------
__global__ void zero_f32(float* __restrict__ a, long na,
                         float* __restrict__ b, long nb) {
  long i = (long)blockIdx.x * blockDim.x + threadIdx.x;
  long stride = (long)gridDim.x * blockDim.x;
  long na4 = na >> 2, nb4 = nb >> 2;
  float4 z = make_float4(0.f, 0.f, 0.f, 0.f);
  for (long j = i; j < na4; j += stride) ((float4*)a)[j] = z;
  for (long j = i; j < nb4; j += stride) ((float4*)b)[j] = z;
  for (long j = (na4 << 2) + i; j < na; j += stride) a[j] = 0.f;
  for (long j = (nb4 << 2) + i; j < nb; j += stride) b[j] = 0.f;
}

// ---------------------------------------------------------------------------
// Per-edge projection (msg @ W) with V_WMMA_F32_16X16X4_F32, fused with
// scatter-add into the stacked output out[node][slot][0..127] and per-node
// edge counting. One block = 64 edges; wave j owns output columns
// j*16..j*16+15 and computes 4 M-subtiles (16x16 each) over K=128.
// ---------------------------------------------------------------------------
__global__ __launch_bounds__(TPB) void proj_scatter_wmma(
    const float* __restrict__ msg, const float* __restrict__ W,
    const int* __restrict__ ei, float* __restrict__ out,
    float* __restrict__ cnt, int slot, int E) {
  __shared__ float sA[TM][LDA];
  __shared__ int sDst[TM];

  const int tid  = threadIdx.x;
  const int lane = tid & 31;
  const int wave = tid >> 5;
  const int e0   = blockIdx.x * TM;

  // Cooperative, coalesced load of the 64x128 fp32 msg tile into LDS.
  for (int i = tid; i < TM * 32; i += TPB) {
    int r  = i >> 5;       // row 0..63
    int c4 = i & 31;       // float4 column 0..31
    float4 v = make_float4(0.f, 0.f, 0.f, 0.f);
    if (e0 + r < E) v = ((const float4*)(msg + (size_t)(e0 + r) * 128))[c4];
    sA[r][c4 * 4 + 0] = v.x; sA[r][c4 * 4 + 1] = v.y;
    sA[r][c4 * 4 + 2] = v.z; sA[r][c4 * 4 + 3] = v.w;
  }
  if (tid < TM) sDst[tid] = (e0 + tid < E) ? ei[2 * (e0 + tid) + 1] : -1;
  __syncthreads();

  // B-fragment layout for 16x16x4 f32 (4x16 K-major): lane L -> N = L%16,
  // K = 2*(L/16) + {0,1} within each K-step of 4. Preload all 32 steps
  // (wave-private 16 columns of W) into registers; W is L2/WGP$-resident.
  const int n     = (wave << 4) + (lane & 15);
  const int khalf = (lane >> 4) << 1;

  v2f bfrag[32];
#pragma unroll
  for (int s = 0; s < 32; ++s) {
    int k = 4 * s + khalf;
    bfrag[s].x = W[(k + 0) * 128 + n];
    bfrag[s].y = W[(k + 1) * 128 + n];
  }

  v8f acc[4] = {v8f{}, v8f{}, v8f{}, v8f{}};
  const int ml = lane & 15;
#pragma unroll
  for (int s = 0; s < 32; ++s) {
    const int kb = 4 * s + khalf;            // even -> 8B-aligned ds_load_b64
    v2f a0 = *(const v2f*)&sA[ 0 + ml][kb];
    v2f a1 = *(const v2f*)&sA[16 + ml][kb];
    v2f a2 = *(const v2f*)&sA[32 + ml][kb];
    v2f a3 = *(const v2f*)&sA[48 + ml][kb];
    acc[0] = __builtin_amdgcn_wmma_f32_16x16x4_f32(false, a0, false, bfrag[s],
                                                   (short)0, acc[0], false, false);
    acc[1] = __builtin_amdgcn_wmma_f32_16x16x4_f32(false, a1, false, bfrag[s],
                                                   (short)0, acc[1], false, false);
    acc[2] = __builtin_amdgcn_wmma_f32_16x16x4_f32(false, a2, false, bfrag[s],
                                                   (short)0, acc[2], false, false);
    acc[3] = __builtin_amdgcn_wmma_f32_16x16x4_f32(false, a3, false, bfrag[s],
                                                   (short)0, acc[3], false, false);
  }

  // C/D layout: lane L, VGPR r -> row M = r + 8*(L/16), col N = L%16.
  // Scatter-add straight into the final stacked layout; lanes 0-15 of a
  // given r hit 16 consecutive floats (one 64B line).
  const int col   = (wave << 4) + (lane & 15);
  const int rbase = (lane >> 4) << 3;
#pragma unroll
  for (int mt = 0; mt < 4; ++mt) {
#pragma unroll
    for (int r = 0; r < 8; ++r) {
      int m   = (mt << 4) + rbase + r;
      int dst = sDst[m];
      if (dst >= 0)
        atomicAdd(out + ((size_t)dst * 2 + slot) * 128 + col, acc[mt][r]);
    }
  }
  if (tid < TM) {
    int dst = sDst[tid];
    if (dst >= 0) atomicAdd(cnt + dst, 1.0f);
  }
}

// ---------------------------------------------------------------------------
// One wave per node: mean = sum / max(cnt,1), nonzero-row masks, num_types.
// ---------------------------------------------------------------------------
__global__ __launch_bounds__(256) void finalize_nodes(
    float* __restrict__ out6, float* __restrict__ out8,
    float* __restrict__ nt6, float* __restrict__ nt8,
    const float* __restrict__ cnt, int N6, int N8) {
  const int wave = threadIdx.x >> 5;
  const int lane = threadIdx.x & 31;
  long gw = (long)blockIdx.x * 8 + wave;

  float* outp; float* ntp; const float* cntp; long node; long Nt;
  if (gw < N6) {
    outp = out6; ntp = nt6; cntp = cnt;                 node = gw;      Nt = N6;
  } else if (gw < (long)N6 + N8) {
    outp = out8; ntp = nt8; cntp = cnt + 2 * (size_t)N6; node = gw - N6; Nt = N8;
  } else {
    return;
  }

  float msum = 0.f;
#pragma unroll
  for (int s = 0; s < 2; ++s) {
    float c   = cntp[(size_t)s * Nt + node];
    float inv = 1.0f / fmaxf(c, 1.0f);
    float* p  = outp + ((size_t)node * 2 + s) * 128;
    float nz  = 0.f;
#pragma unroll
    for (int q = 0; q < 4; ++q) {
      float v = p[lane + (q << 5)] * inv;
      p[lane + (q << 5)] = v;
      if (v != 0.f) nz = 1.f;
    }
    for (int off = 16; off > 0; off >>= 1) nz += __shfl_down(nz, off);
    nz = __shfl(nz, 0);
    msum += (nz > 0.f) ? 1.f : 0.f;
  }
  if (lane == 0) ntp[node] = msum;
}

// ---------------------------------------------------------------------------
extern "C" void kernel_launch(void* const* d_in, const int* in_sizes, int n_in,
                              void* d_out, int out_size, void* d_ws, size_t ws_size,
                              hipStream_t stream) {
  const float* msg66 = (const float*)d_in[0];
  const float* msg68 = (const float*)d_in[1];
  const float* msg86 = (const float*)d_in[2];
  const float* msg88 = (const float*)d_in[3];
  const float* W66   = (const float*)d_in[4];
  const float* W68   = (const float*)d_in[5];
  const float* W86   = (const float*)d_in[6];
  const float* W88   = (const float*)d_in[7];
  const int*   ei66  = (const int*)d_in[8];
  const int*   ei68  = (const int*)d_in[9];
  const int*   ei86  = (const int*)d_in[10];
  const int*   ei88  = (const int*)d_in[11];

  const int N6 = 100000, N8 = 100000;     // fixed in the reference
  const int E  = in_sizes[0] / 128;       // 400000

  float* out6 = (float*)d_out;
  float* out8 = out6 + (size_t)N6 * 2 * 128;
  float* nt6  = out8 + (size_t)N8 * 2 * 128;
  float* nt8  = nt6 + N6;
  float* cnt  = (float*)d_ws;             // [c66|c86|c68|c88] = 2*N6+2*N8 floats

  zero_f32<<<2048, 256, 0, stream>>>((float*)d_out, (long)out_size,
                                     cnt, 2L * N6 + 2L * N8);

  const int gb = (E + TM - 1) / TM;       // 6250 blocks per edge type
  proj_scatter_wmma<<<gb, TPB, 0, stream>>>(msg66, W66, ei66, out6, cnt,            0, E);
  proj_scatter_wmma<<<gb, TPB, 0, stream>>>(msg86, W86, ei86, out6, cnt + N6,       1, E);
  proj_scatter_wmma<<<gb, TPB, 0, stream>>>(msg68, W68, ei68, out8, cnt + 2*N6,     0, E);
  proj_scatter_wmma<<<gb, TPB, 0, stream>>>(msg88, W88, ei88, out8, cnt + 2*N6+N8,  1, E);

  const int totWaves = N6 + N8;
  finalize_nodes<<<(totWaves + 7) / 8, 256, 0, stream>>>(out6, out8, nt6, nt8,
                                                         cnt, N6, N8);
}